// Structure2VecLayer_41162966565589
// MI455X (gfx1250) — compile-verified
//
#include <hip/hip_runtime.h>

// Structure2Vec layer for MI455X (gfx1250, wave32).
//
// Strategy:
//  - scatter: acc1[dst] += features[src] (f32 atomics, L2-resident 25.6MB acc),
//             accw[dst] += edge_w (16 floats/edge), cnt[dst] += 1
//    (uses segment_sum(edge_w@bond_W+b) == segment_sum(edge_w)@bond_W + cnt*b)
//  - gemm1: y = relu(acc1@W1 + accw@bond_W + b1 + cnt*bond_b), via
//           V_WMMA_F32_16X16X4_F32, fused column sum/sumsq for BN1
//  - bn_stats: scale/shift from sums
//  - gemm2: out = relu((y*scale1+shift1)@W2 + b2 + features), fused BN2 stats
//  - bn_apply: out = out*scale2 + shift2  (in place)

#define N_NODES 50000
#define N_EDGES 800000
#define H 128
#define FB 16
#define BN_EPS 1e-5f

typedef __attribute__((ext_vector_type(2))) float v2f;
typedef __attribute__((ext_vector_type(8))) float v8f;

__device__ __forceinline__ v8f wmma_f32(v2f a, v2f b, v8f c) {
  // D = A(16x4) * B(4x16) + C(16x16), all f32, wave32
  return __builtin_amdgcn_wmma_f32_16x16x4_f32(
      /*neg_a=*/false, a, /*neg_b=*/false, b,
      /*c_mod=*/(short)0, c, /*reuse_a=*/false, /*reuse_b=*/false);
}

__global__ void zero_kernel(float* __restrict__ p, long n) {
  long i = (long)blockIdx.x * blockDim.x + threadIdx.x;
  long stride = (long)gridDim.x * blockDim.x;
  for (; i < n; i += stride) p[i] = 0.0f;
}

// 32 lanes per edge: each lane moves 4 columns of features[src] -> acc1[dst],
// lanes 0..15 also scatter edge_w, lane 31 bumps the destination count.
__global__ void scatter_kernel(const float* __restrict__ feat,
                               const float* __restrict__ edge_w,
                               const int* __restrict__ src,
                               const int* __restrict__ dst,
                               float* __restrict__ acc1,
                               float* __restrict__ accw,
                               float* __restrict__ cnt) {
  int tid = blockIdx.x * blockDim.x + threadIdx.x;
  int e = tid >> 5;
  int lane = tid & 31;
  if (e >= N_EDGES) return;
  int s = src[e];
  int d = dst[e];
  const float4* frow = (const float4*)(feat + (size_t)s * H);
  float4 v = frow[lane];                       // columns lane*4 .. lane*4+3
  float* arow = acc1 + (size_t)d * H + lane * 4;
  atomicAdd(arow + 0, v.x);
  atomicAdd(arow + 1, v.y);
  atomicAdd(arow + 2, v.z);
  atomicAdd(arow + 3, v.w);
  if (lane < FB)
    atomicAdd(accw + (size_t)d * FB + lane, edge_w[(size_t)e * FB + lane]);
  if (lane == 31) atomicAdd(cnt + d, 1.0f);
}

// One block = 16 rows x 128 cols. 4 waves, each wave owns two 16x16 col tiles.
// K = 128 (A@W1) + 16 (accw@bond_W) as f32 WMMA steps of K=4.
// Epilogue: +b1 +cnt*bond_b, ReLU, in-place store to Y(=acc1), BN stats.
__global__ void gemm1_kernel(const float* __restrict__ A,      // [N,H] acc1
                             const float* __restrict__ W1,     // [H,H]
                             const float* __restrict__ bias1,  // [H]
                             const float* __restrict__ accw,   // [N,FB]
                             const float* __restrict__ bondW,  // [FB,H]
                             const float* __restrict__ bondb,  // [H]
                             const float* __restrict__ cnt,    // [N]
                             float* __restrict__ Y,            // [N,H] (=acc1)
                             float* __restrict__ colsum,
                             float* __restrict__ colsumsq) {
  __shared__ float s_sum[H];
  __shared__ float s_sq[H];
  int tid = threadIdx.x;          // 0..127 (4 wave32)
  int wave = tid >> 5;
  int lane = tid & 31;
  int rows0 = blockIdx.x * 16;
  int mA = lane & 15;             // A-matrix row for this lane
  int khalf = (lane >> 4) << 1;   // lanes 16-31 hold K+2,K+3

  if (tid < H) { s_sum[tid] = 0.0f; s_sq[tid] = 0.0f; }
  __syncthreads();

  const float* arow = A + (size_t)(rows0 + mA) * H;
  const float* wrow = accw + (size_t)(rows0 + mA) * FB;
  int n0 = wave * 32 + (lane & 15);
  int n1 = n0 + 16;

  v8f c0 = {};
  v8f c1 = {};
  for (int k0 = 0; k0 < H; k0 += 4) {
    int kb = k0 + khalf;
    v2f a = {arow[kb], arow[kb + 1]};
    v2f bA = {W1[kb * H + n0], W1[(kb + 1) * H + n0]};
    v2f bB = {W1[kb * H + n1], W1[(kb + 1) * H + n1]};
    c0 = wmma_f32(a, bA, c0);
    c1 = wmma_f32(a, bB, c1);
  }
  for (int k0 = 0; k0 < FB; k0 += 4) {   // fused edge-feature projection
    int kb = k0 + khalf;
    v2f a = {wrow[kb], wrow[kb + 1]};
    v2f bA = {bondW[kb * H + n0], bondW[(kb + 1) * H + n0]};
    v2f bB = {bondW[kb * H + n1], bondW[(kb + 1) * H + n1]};
    c0 = wmma_f32(a, bA, c0);
    c1 = wmma_f32(a, bB, c1);
  }

  __syncthreads();  // all reads of A done before the in-place store below

  int mbase = (lane >> 4) << 3;  // C/D layout: lanes>=16 hold M=8..15
  float bi0 = bias1[n0], bi1 = bias1[n1];
  float bo0 = bondb[n0], bo1 = bondb[n1];
  float ps0 = 0.f, pq0 = 0.f, ps1 = 0.f, pq1 = 0.f;
#pragma unroll
  for (int i = 0; i < 8; ++i) {
    int m = rows0 + mbase + i;
    float cn = cnt[m];
    float v0 = fmaxf(c0[i] + bi0 + cn * bo0, 0.0f);
    float v1 = fmaxf(c1[i] + bi1 + cn * bo1, 0.0f);
    Y[(size_t)m * H + n0] = v0;
    Y[(size_t)m * H + n1] = v1;
    ps0 += v0; pq0 += v0 * v0;
    ps1 += v1; pq1 += v1 * v1;
  }
  ps0 += __shfl_xor(ps0, 16, 32);
  pq0 += __shfl_xor(pq0, 16, 32);
  ps1 += __shfl_xor(ps1, 16, 32);
  pq1 += __shfl_xor(pq1, 16, 32);
  if (lane < 16) {
    atomicAdd(&s_sum[n0], ps0);
    atomicAdd(&s_sq[n0], pq0);
    atomicAdd(&s_sum[n1], ps1);
    atomicAdd(&s_sq[n1], pq1);
  }
  __syncthreads();
  if (tid < H) {
    atomicAdd(&colsum[tid], s_sum[tid]);
    atomicAdd(&colsumsq[tid], s_sq[tid]);
  }
}

// out = relu((Y*scale1+shift1) @ W2 + b2 + features), fused BN2 stats.
__global__ void gemm2_kernel(const float* __restrict__ Y,       // [N,H]
                             const float* __restrict__ scale1,  // [H]
                             const float* __restrict__ shift1,  // [H]
                             const float* __restrict__ W2,      // [H,H]
                             const float* __restrict__ bias2,   // [H]
                             const float* __restrict__ feat,    // [N,H]
                             float* __restrict__ OUT,           // [N,H]
                             float* __restrict__ colsum,
                             float* __restrict__ colsumsq) {
  __shared__ float s_sum[H];
  __shared__ float s_sq[H];
  int tid = threadIdx.x;
  int wave = tid >> 5;
  int lane = tid & 31;
  int rows0 = blockIdx.x * 16;
  int mA = lane & 15;
  int khalf = (lane >> 4) << 1;

  if (tid < H) { s_sum[tid] = 0.0f; s_sq[tid] = 0.0f; }
  __syncthreads();

  const float* yrow = Y + (size_t)(rows0 + mA) * H;
  int n0 = wave * 32 + (lane & 15);
  int n1 = n0 + 16;

  v8f c0 = {};
  v8f c1 = {};
  for (int k0 = 0; k0 < H; k0 += 4) {
    int kb = k0 + khalf;
    // apply BN1 affine on the fly while loading A
    v2f a = {yrow[kb] * scale1[kb] + shift1[kb],
             yrow[kb + 1] * scale1[kb + 1] + shift1[kb + 1]};
    v2f bA = {W2[kb * H + n0], W2[(kb + 1) * H + n0]};
    v2f bB = {W2[kb * H + n1], W2[(kb + 1) * H + n1]};
    c0 = wmma_f32(a, bA, c0);
    c1 = wmma_f32(a, bB, c1);
  }

  int mbase = (lane >> 4) << 3;
  float bi0 = bias2[n0], bi1 = bias2[n1];
  float ps0 = 0.f, pq0 = 0.f, ps1 = 0.f, pq1 = 0.f;
#pragma unroll
  for (int i = 0; i < 8; ++i) {
    int m = rows0 + mbase + i;
    const float* frow = feat + (size_t)m * H;
    float v0 = fmaxf(c0[i] + bi0 + frow[n0], 0.0f);
    float v1 = fmaxf(c1[i] + bi1 + frow[n1], 0.0f);
    OUT[(size_t)m * H + n0] = v0;
    OUT[(size_t)m * H + n1] = v1;
    ps0 += v0; pq0 += v0 * v0;
    ps1 += v1; pq1 += v1 * v1;
  }
  ps0 += __shfl_xor(ps0, 16, 32);
  pq0 += __shfl_xor(pq0, 16, 32);
  ps1 += __shfl_xor(ps1, 16, 32);
  pq1 += __shfl_xor(pq1, 16, 32);
  if (lane < 16) {
    atomicAdd(&s_sum[n0], ps0);
    atomicAdd(&s_sq[n0], pq0);
    atomicAdd(&s_sum[n1], ps1);
    atomicAdd(&s_sq[n1], pq1);
  }
  __syncthreads();
  if (tid < H) {
    atomicAdd(&colsum[tid], s_sum[tid]);
    atomicAdd(&colsumsq[tid], s_sq[tid]);
  }
}

__global__ void bn_stats_kernel(const float* __restrict__ colsum,
                                const float* __restrict__ colsumsq,
                                const float* __restrict__ gamma,
                                const float* __restrict__ beta,
                                float* __restrict__ scale,
                                float* __restrict__ shift) {
  int i = threadIdx.x;  // one block, H threads
  const float invN = 1.0f / (float)N_NODES;
  float mean = colsum[i] * invN;
  float var = colsumsq[i] * invN - mean * mean;
  float inv = rsqrtf(var + BN_EPS);
  float sc = gamma[i] * inv;
  scale[i] = sc;
  shift[i] = beta[i] - mean * sc;
}

__global__ void bn_apply_kernel(float* __restrict__ out,
                                const float* __restrict__ scale,
                                const float* __restrict__ shift) {
  long i = (long)blockIdx.x * blockDim.x + threadIdx.x;
  if (i < (long)N_NODES * H) {
    int col = (int)(i & (H - 1));
    out[i] = out[i] * scale[col] + shift[col];
  }
}

extern "C" void kernel_launch(void* const* d_in, const int* in_sizes, int n_in,
                              void* d_out, int out_size, void* d_ws,
                              size_t ws_size, hipStream_t stream) {
  (void)in_sizes; (void)n_in; (void)out_size; (void)ws_size;
  const float* feat   = (const float*)d_in[0];
  const float* edge_w = (const float*)d_in[1];
  const int*   src    = (const int*)d_in[2];
  const int*   dst    = (const int*)d_in[3];
  const float* bondW  = (const float*)d_in[4];
  const float* bondb  = (const float*)d_in[5];
  const float* W1     = (const float*)d_in[6];
  const float* b1     = (const float*)d_in[7];
  const float* W2     = (const float*)d_in[8];
  const float* b2     = (const float*)d_in[9];
  const float* gamma1 = (const float*)d_in[10];
  const float* beta1  = (const float*)d_in[11];
  const float* gamma2 = (const float*)d_in[12];
  const float* beta2  = (const float*)d_in[13];
  float* out = (float*)d_out;
  float* ws = (float*)d_ws;

  // workspace layout (floats)
  float* acc1 = ws;                               // N*H  (also Y, in place)
  float* accw = acc1 + (size_t)N_NODES * H;       // N*FB
  float* cnt = accw + (size_t)N_NODES * FB;       // N
  float* colsum1 = cnt + N_NODES;                 // H
  float* colsumsq1 = colsum1 + H;                 // H
  float* colsum2 = colsumsq1 + H;                 // H
  float* colsumsq2 = colsum2 + H;                 // H
  float* scale1 = colsumsq2 + H;                  // H
  float* shift1 = scale1 + H;                     // H
  float* scale2 = shift1 + H;                     // H
  float* shift2 = scale2 + H;                     // H

  long zero_n = (long)N_NODES * H + (long)N_NODES * FB + N_NODES + 4 * H;
  int zblocks = (int)((zero_n + 255) / 256);
  zero_kernel<<<zblocks, 256, 0, stream>>>(ws, zero_n);

  scatter_kernel<<<(N_EDGES * 32) / 256, 256, 0, stream>>>(
      feat, edge_w, src, dst, acc1, accw, cnt);

  gemm1_kernel<<<N_NODES / 16, 128, 0, stream>>>(
      acc1, W1, b1, accw, bondW, bondb, cnt, acc1, colsum1, colsumsq1);

  bn_stats_kernel<<<1, H, 0, stream>>>(colsum1, colsumsq1, gamma1, beta1,
                                       scale1, shift1);

  gemm2_kernel<<<N_NODES / 16, 128, 0, stream>>>(
      acc1, scale1, shift1, W2, b2, feat, out, colsum2, colsumsq2);

  bn_stats_kernel<<<1, H, 0, stream>>>(colsum2, colsumsq2, gamma2, beta2,
                                       scale2, shift2);

  bn_apply_kernel<<<(N_NODES * H) / 256, 256, 0, stream>>>(out, scale2, shift2);
}